// GatedGCNLSPE_27900107555157
// MI455X (gfx1250) — compile-verified
//
#include <hip/hip_runtime.h>

// ---------------------------------------------------------------------------
// GatedGCN-LSPE on MI455X (gfx1250), fp32 WMMA (16x16x4), wave32.
// Round 3: K-pair-interleaved LDS layout for W so each B fragment is one
//          aligned ds_load_b64 (no v_mov packing); bank-conflict-free stride.
// ---------------------------------------------------------------------------

typedef float v2f __attribute__((ext_vector_type(2)));
typedef float v8f __attribute__((ext_vector_type(8)));

#define DD 96          // hidden dim (fixed by the model)
#define NTILE 6        // 96 / 16 output column tiles
#define TPB 256        // 8 wave32 per block
#define ROWS_PER_WAVE 32
#define ROWS_PER_BLOCK 256     // 8 waves * 32 rows
#define PAIR_STRIDE (2 * DD + 32)  // floats per K-pair row; 224 ≡ 32 banks (mod 64)

static __device__ __forceinline__ v8f wmma_f32(v2f a, v2f b, v8f c) {
  // V_WMMA_F32_16X16X4_F32: (neg_a, A, neg_b, B, c_mod, C, reuse_a, reuse_b)
  return __builtin_amdgcn_wmma_f32_16x16x4_f32(false, a, false, b, (short)0, c,
                                               false, false);
}

static __device__ __forceinline__ long clampRow(long r, long M) {
  return r < M ? r : (M - 1);
}

// Stage W[K,96] (row-major) into LDS with K-pair interleave:
//   sW[(k>>1)*PAIR_STRIDE + n*2 + (k&1)] = W[k*96 + n]
static __device__ __forceinline__ void stageW(float* sW,
                                              const float* __restrict__ W,
                                              int K) {
  for (int i = threadIdx.x; i < K * DD; i += TPB) {
    int k = i / DD;
    int n = i - k * DD;
    sW[(k >> 1) * PAIR_STRIDE + n * 2 + (k & 1)] = W[i];
  }
  __syncthreads();
}

// ---------------------------------------------------------------------------
// Generic GEMM: C[M,96] = A[M,K] @ W[K,96]  (+bias) (+=C) (relu)
// flags: bit0 = add bias, bit1 = accumulate into existing C, bit2 = relu
// One wave computes a 32x96 strip. Loads clamped (branch-free, EXEC stays
// full for WMMA); stores guarded.
// ---------------------------------------------------------------------------
__global__ void gemm96(const float* __restrict__ A, const float* __restrict__ W,
                       const float* __restrict__ bias, float* __restrict__ C,
                       long M, int K, int flags) {
  extern __shared__ float sW[];
  stageW(sW, W, K);

  const int wv = threadIdx.x >> 5;
  const int lane = threadIdx.x & 31;
  const long m0 = (long)blockIdx.x * ROWS_PER_BLOCK + (long)wv * ROWS_PER_WAVE;
  if (m0 >= M) return;
  const int half = lane >> 4;
  const int nlo = lane & 15;

  const float* Ap0 = A + clampRow(m0 + nlo, M) * (long)K;
  const float* Ap1 = A + clampRow(m0 + 16 + nlo, M) * (long)K;

  v8f acc[2][NTILE];
  if (flags & 2) {
#pragma unroll
    for (int m = 0; m < 2; ++m)
#pragma unroll
      for (int t = 0; t < NTILE; ++t)
#pragma unroll
        for (int r = 0; r < 8; ++r) {
          long row = clampRow(m0 + 16 * m + r + 8 * half, M);
          acc[m][t][r] = C[row * DD + t * 16 + nlo];
        }
  } else {
#pragma unroll
    for (int m = 0; m < 2; ++m)
#pragma unroll
      for (int t = 0; t < NTILE; ++t)
#pragma unroll
        for (int r = 0; r < 8; ++r) acc[m][t][r] = 0.f;
  }

  for (int kk = 0; kk < K; kk += 4) {
    const int kb = kk + 2 * half;
    v2f a0 = *(const v2f*)(Ap0 + kb);
    v2f a1 = *(const v2f*)(Ap1 + kb);
    const float* bp = sW + (kk / 2 + half) * PAIR_STRIDE + nlo * 2;
#pragma unroll
    for (int t = 0; t < NTILE; ++t) {
      v2f b = *(const v2f*)(bp + t * 32);  // {W[kb][n], W[kb+1][n]} contiguous
      acc[0][t] = wmma_f32(a0, b, acc[0][t]);
      acc[1][t] = wmma_f32(a1, b, acc[1][t]);
    }
  }

#pragma unroll
  for (int m = 0; m < 2; ++m)
#pragma unroll
    for (int t = 0; t < NTILE; ++t)
#pragma unroll
      for (int r = 0; r < 8; ++r) {
        long row = m0 + 16 * m + r + 8 * half;
        if (row < M) {
          int col = t * 16 + nlo;
          float v = acc[m][t][r];
          if (flags & 1) v += bias[col];
          if (flags & 4) v = fmaxf(v, 0.f);
          C[row * DD + col] = v;
        }
      }
}

// ---------------------------------------------------------------------------
// Gate GEMM over edges with fused gather epilogue:
//   eta_hat[e,:] = sigmoid(e_state[e,:]@Wg_e + As[send[e],:] + Ar[rec[e],:] + b)
// ---------------------------------------------------------------------------
__global__ void gemm96_gate(const float* __restrict__ Est,
                            const float* __restrict__ W,
                            const float* __restrict__ gb,
                            const float* __restrict__ As,
                            const float* __restrict__ Ar,
                            const int* __restrict__ send,
                            const int* __restrict__ rec,
                            float* __restrict__ eta_hat, long E) {
  extern __shared__ float sW[];
  stageW(sW, W, DD);

  const int wv = threadIdx.x >> 5;
  const int lane = threadIdx.x & 31;
  const long m0 = (long)blockIdx.x * ROWS_PER_BLOCK + (long)wv * ROWS_PER_WAVE;
  if (m0 >= E) return;
  const int half = lane >> 4;
  const int nlo = lane & 15;

  const float* Ap0 = Est + clampRow(m0 + nlo, E) * (long)DD;
  const float* Ap1 = Est + clampRow(m0 + 16 + nlo, E) * (long)DD;

  v8f acc[2][NTILE];
#pragma unroll
  for (int m = 0; m < 2; ++m)
#pragma unroll
    for (int t = 0; t < NTILE; ++t)
#pragma unroll
      for (int r = 0; r < 8; ++r) acc[m][t][r] = 0.f;

  for (int kk = 0; kk < DD; kk += 4) {
    const int kb = kk + 2 * half;
    v2f a0 = *(const v2f*)(Ap0 + kb);
    v2f a1 = *(const v2f*)(Ap1 + kb);
    const float* bp = sW + (kk / 2 + half) * PAIR_STRIDE + nlo * 2;
#pragma unroll
    for (int t = 0; t < NTILE; ++t) {
      v2f b = *(const v2f*)(bp + t * 32);
      acc[0][t] = wmma_f32(a0, b, acc[0][t]);
      acc[1][t] = wmma_f32(a1, b, acc[1][t]);
    }
  }

#pragma unroll
  for (int m = 0; m < 2; ++m)
#pragma unroll
    for (int r = 0; r < 8; ++r) {
      long row = m0 + 16 * m + r + 8 * half;
      long rl = clampRow(row, E);
      long s = send[rl], rr = rec[rl];
      if (row < E) {
#pragma unroll
        for (int t = 0; t < NTILE; ++t) {
          int col = t * 16 + nlo;
          float v = acc[m][t][r] + gb[col] + As[s * DD + col] + Ar[rr * DD + col];
          eta_hat[row * DD + col] = 1.f / (1.f + __expf(-v));
        }
      }
    }
}

// ---------------------------------------------------------------------------
// Edge message kernel: one wave32 per edge.
//   eta = eta_hat / rowsum;  h_agg[rec] += eta*nhps[send];
//   p_agg[rec] += eta*nps[send];  e_state += relu(eta_hat)
// ---------------------------------------------------------------------------
__global__ void edge_msg(const float* __restrict__ eta_hat,
                         const int* __restrict__ send,
                         const int* __restrict__ rec,
                         const float* __restrict__ nhps,
                         const float* __restrict__ nps,
                         float* __restrict__ e_state,
                         float* __restrict__ h_agg,
                         float* __restrict__ p_agg, long E) {
  long wid = ((long)blockIdx.x * blockDim.x + threadIdx.x) >> 5;
  int lane = threadIdx.x & 31;
  if (wid >= E) return;
  const float* row = eta_hat + wid * DD;
  float v0 = row[lane], v1 = row[lane + 32], v2 = row[lane + 64];
  float s = v0 + v1 + v2;
#pragma unroll
  for (int off = 16; off > 0; off >>= 1) s += __shfl_xor(s, off, 32);
  float inv = 1.f / s;
  long snd = send[wid], rc = rec[wid];
  const float* hs = nhps + snd * DD;
  const float* ps = nps + snd * DD;
  float* ha = h_agg + rc * DD;
  float* pa = p_agg + rc * DD;
  float* er = e_state + wid * DD;
#pragma unroll
  for (int c = 0; c < 3; ++c) {
    int col = lane + 32 * c;
    float eh = (c == 0) ? v0 : ((c == 1) ? v1 : v2);
    float eta = eh * inv;
    atomicAdd(&ha[col], eta * hs[col]);
    atomicAdd(&pa[col], eta * ps[col]);
    er[col] += fmaxf(eh, 0.f);
  }
}

// ---------------------------------------------------------------------------
// Elementwise / pooling / readout helpers
// ---------------------------------------------------------------------------
__global__ void zero_k(float* __restrict__ p, long n) {
  for (long i = (long)blockIdx.x * blockDim.x + threadIdx.x; i < n;
       i += (long)gridDim.x * blockDim.x)
    p[i] = 0.f;
}

__global__ void add_k(float* __restrict__ dst, const float* __restrict__ src,
                      long n) {
  for (long i = (long)blockIdx.x * blockDim.x + threadIdx.x; i < n;
       i += (long)gridDim.x * blockDim.x)
    dst[i] += src[i];
}

// h += relu(h_new); p += tanh(p_lin + p_agg)
__global__ void final_update(float* __restrict__ h, float* __restrict__ p,
                             const float* __restrict__ h_new,
                             const float* __restrict__ p_lin,
                             const float* __restrict__ p_agg, long n) {
  for (long i = (long)blockIdx.x * blockDim.x + threadIdx.x; i < n;
       i += (long)gridDim.x * blockDim.x) {
    h[i] += fmaxf(h_new[i], 0.f);
    p[i] += tanhf(p_lin[i] + p_agg[i]);
  }
}

__global__ void pool_k(const float* __restrict__ h, const float* __restrict__ p,
                       const int* __restrict__ batch,
                       float* __restrict__ hpool, float* __restrict__ ppool,
                       long n) {
  for (long i = (long)blockIdx.x * blockDim.x + threadIdx.x; i < n;
       i += (long)gridDim.x * blockDim.x) {
    long node = i / DD;
    int c = (int)(i - node * DD);
    long g = batch[node];
    atomicAdd(&hpool[g * DD + c], h[i]);
    atomicAdd(&ppool[g * DD + c], p[i]);
  }
}

__global__ void ro2_k(const float* __restrict__ x1, const float* __restrict__ w2,
                      const float* __restrict__ b2, float* __restrict__ out,
                      int G) {
  int g = blockIdx.x * blockDim.x + threadIdx.x;
  if (g < G) {
    float s = b2[0];
#pragma unroll 8
    for (int j = 0; j < DD; ++j) s += x1[(long)g * DD + j] * w2[j];
    out[g] = s;
  }
}

// ---------------------------------------------------------------------------
// Host orchestration
// ---------------------------------------------------------------------------
static inline int gemm_blocks(long M) {
  return (int)((M + ROWS_PER_BLOCK - 1) / ROWS_PER_BLOCK);
}
static inline int ew_blocks(long n) {
  long b = (n + TPB - 1) / TPB;
  return (int)(b < 8192 ? b : 8192);
}
static inline size_t shW(int K) {  // LDS bytes for pair-interleaved W
  return (size_t)(K / 2) * PAIR_STRIDE * sizeof(float);
}

extern "C" void kernel_launch(void* const* d_in, const int* in_sizes, int n_in,
                              void* d_out, int out_size, void* d_ws,
                              size_t ws_size, hipStream_t stream) {
  const float* h_in = (const float*)d_in[0];
  const float* e_in = (const float*)d_in[1];
  const float* p_in = (const float*)d_in[2];
  const int* eidx = (const int*)d_in[3];
  const int* batch = (const int*)d_in[4];
  const float* h_emb_w = (const float*)d_in[5];
  const float* h_emb_b = (const float*)d_in[6];
  const float* e_emb_w = (const float*)d_in[7];
  const float* e_emb_b = (const float*)d_in[8];
  const float* p_emb_w = (const float*)d_in[9];
  const float* p_emb_b = (const float*)d_in[10];
  const float* gate_w = (const float*)d_in[11];
  const float* gate_b = (const float*)d_in[12];
  const float* hps_w = (const float*)d_in[13];
  const float* hps_b = (const float*)d_in[14];
  const float* hpr_w = (const float*)d_in[15];
  const float* hpr_b = (const float*)d_in[16];
  const float* ps_w = (const float*)d_in[17];
  const float* ps_b = (const float*)d_in[18];
  const float* pr_w = (const float*)d_in[19];
  const float* pr_b = (const float*)d_in[20];
  const float* ro1_w = (const float*)d_in[21];
  const float* ro1_b = (const float*)d_in[22];
  const float* ro2_w = (const float*)d_in[23];
  const float* ro2_b = (const float*)d_in[24];
  float* out = (float*)d_out;

  const int D = in_sizes[6];           // 96
  const int FI = in_sizes[5] / D;      // 128
  const long N = in_sizes[0] / FI;     // 50000
  const int EI = in_sizes[7] / D;      // 16
  const long E = in_sizes[1] / EI;     // 800000
  const int PI = in_sizes[9] / D;      // 16
  const int L = in_sizes[12] / D;      // 4
  const int G = out_size;              // 256

  const int* send = eidx;
  const int* rec = eidx + E;

  // workspace layout (all f32)
  float* w = (float*)d_ws;
  float* h_st = w;  w += N * D;
  float* p_st = w;  w += N * D;
  float* As = w;    w += N * D;
  float* Ar = w;    w += N * D;
  float* nhps = w;  w += N * D;
  float* nhpr = w;  w += N * D;  // later holds h_new
  float* nps = w;   w += N * D;
  float* hagg = w;  w += N * D;
  float* pagg = w;  w += N * D;
  float* plin = w;  w += N * D;
  float* hpool = w; w += (long)G * D;
  float* ppool = w; w += (long)G * D;
  float* x1 = w;    w += (long)G * D;
  float* e_st = w;  w += E * D;
  float* etah = w;  w += E * D;

  const size_t shDD = shW(D);

  // ---- input embeddings ----
  gemm96<<<gemm_blocks(N), TPB, shW(FI), stream>>>(h_in, h_emb_w, h_emb_b, h_st,
                                                   N, FI, 1);
  gemm96<<<gemm_blocks(E), TPB, shW(EI), stream>>>(e_in, e_emb_w, e_emb_b, e_st,
                                                   E, EI, 1);
  gemm96<<<gemm_blocks(N), TPB, shW(PI), stream>>>(p_in, p_emb_w, p_emb_b, p_st,
                                                   N, PI, 1);

  // ---- layers ----
  for (int l = 0; l < L; ++l) {
    const float* Wg = gate_w + (size_t)l * 3 * D * D;
    const float* Bg = gate_b + (size_t)l * D;
    const float* Whps = hps_w + (size_t)l * 2 * D * D;
    const float* Bhps = hps_b + (size_t)l * D;
    const float* Whpr = hpr_w + (size_t)l * 2 * D * D;
    const float* Bhpr = hpr_b + (size_t)l * D;
    const float* Wps = ps_w + (size_t)l * D * D;
    const float* Bps = ps_b + (size_t)l * D;
    const float* Wpr = pr_w + (size_t)l * D * D;
    const float* Bpr = pr_b + (size_t)l * D;

    // node-level gate projections (factored out of the edge concat)
    gemm96<<<gemm_blocks(N), TPB, shDD, stream>>>(h_st, Wg, nullptr, As, N, D, 0);
    gemm96<<<gemm_blocks(N), TPB, shDD, stream>>>(h_st, Wg + D * D, nullptr, Ar, N, D, 0);
    // edge GEMM + gather + sigmoid
    gemm96_gate<<<gemm_blocks(E), TPB, shDD, stream>>>(
        e_st, Wg + 2 * D * D, Bg, As, Ar, send, rec, etah, E);

    // node projections (concat as two accumulating GEMMs)
    gemm96<<<gemm_blocks(N), TPB, shDD, stream>>>(h_st, Whps, nullptr, nhps, N, D, 0);
    gemm96<<<gemm_blocks(N), TPB, shDD, stream>>>(p_st, Whps + D * D, Bhps, nhps, N, D, 3);
    gemm96<<<gemm_blocks(N), TPB, shDD, stream>>>(h_st, Whpr, nullptr, nhpr, N, D, 0);
    gemm96<<<gemm_blocks(N), TPB, shDD, stream>>>(p_st, Whpr + D * D, Bhpr, nhpr, N, D, 3);
    gemm96<<<gemm_blocks(N), TPB, shDD, stream>>>(p_st, Wps, Bps, nps, N, D, 1);

    zero_k<<<ew_blocks(N * D), TPB, 0, stream>>>(hagg, N * D);
    zero_k<<<ew_blocks(N * D), TPB, 0, stream>>>(pagg, N * D);

    edge_msg<<<(int)((E * 32 + TPB - 1) / TPB), TPB, 0, stream>>>(
        etah, send, rec, nhps, nps, e_st, hagg, pagg, E);

    // h_new = hpr(node) + h_agg  (in place into nhpr)
    add_k<<<ew_blocks(N * D), TPB, 0, stream>>>(nhpr, hagg, N * D);
    // p_lin = h_new @ pr_w + pr_b
    gemm96<<<gemm_blocks(N), TPB, shDD, stream>>>(nhpr, Wpr, Bpr, plin, N, D, 1);
    // h += relu(h_new); p += tanh(p_lin + p_agg)
    final_update<<<ew_blocks(N * D), TPB, 0, stream>>>(h_st, p_st, nhpr, plin,
                                                       pagg, N * D);
  }

  // ---- pooling + readout ----
  zero_k<<<ew_blocks((long)G * D), TPB, 0, stream>>>(hpool, (long)G * D);
  zero_k<<<ew_blocks((long)G * D), TPB, 0, stream>>>(ppool, (long)G * D);
  pool_k<<<ew_blocks(N * D), TPB, 0, stream>>>(h_st, p_st, batch, hpool, ppool,
                                               N * D);
  gemm96<<<gemm_blocks(G), TPB, shDD, stream>>>(hpool, ro1_w, nullptr, x1, G, D, 0);
  gemm96<<<gemm_blocks(G), TPB, shDD, stream>>>(ppool, ro1_w + D * D, ro1_b, x1,
                                                G, D, 1 | 2 | 4);
  ro2_k<<<(G + 63) / 64, 64, 0, stream>>>(x1, ro2_w, ro2_b, out, G);
}